// struct2vec_model_29248727286147
// MI455X (gfx1250) — compile-verified
//
#include <hip/hip_runtime.h>
#include <hip/hip_bf16.h>

#define N_NEI 64
#define Bk 8192
#define Pk 128

typedef __attribute__((ext_vector_type(2))) float v2f;
typedef __attribute__((ext_vector_type(8))) float v8f;

// ---------------------------------------------------------------------------
// Kernel 1: S[b,p] = sum_n mu_N[n,b,p]   (256 MB stream -> bandwidth bound)
// ---------------------------------------------------------------------------
__global__ __launch_bounds__(256) void reduce_mu(const float* __restrict__ muN,
                                                 float* __restrict__ S) {
    const int idx = blockIdx.x * 256 + threadIdx.x;      // float4 index, [0, B*P/4)
    const float4* src = (const float4*)muN;
    const int stride4 = (Bk * Pk) / 4;                   // 262144 float4 per neighbor slice
    float4 acc = make_float4(0.f, 0.f, 0.f, 0.f);
#pragma unroll 8
    for (int n = 0; n < N_NEI; ++n) {
        float4 v = src[(size_t)n * stride4 + idx];
        acc.x += v.x; acc.y += v.y; acc.z += v.z; acc.w += v.w;
    }
    ((float4*)S)[idx] = acc;
}

// ---------------------------------------------------------------------------
// Kernel 2: per-b signed sums of wi/ui/ti over the neighbor axis
//   sc[0*B+b]=sum max(wi,0)  sc[1*B+b]=sum min(wi,0)  (then ui, then ti)
// ---------------------------------------------------------------------------
__global__ __launch_bounds__(256) void reduce_wut(const float* __restrict__ wi,
                                                  const float* __restrict__ ui,
                                                  const float* __restrict__ ti,
                                                  float* __restrict__ sc) {
    const int b = blockIdx.x * 256 + threadIdx.x;        // [0, B)
    float pw = 0.f, nw = 0.f, pu = 0.f, nu = 0.f, pt = 0.f, nt = 0.f;
#pragma unroll 4
    for (int n = 0; n < N_NEI; ++n) {
        float w = wi[n * Bk + b];
        float u = ui[n * Bk + b];
        float t = ti[n * Bk + b];
        pw += fmaxf(w, 0.f); nw += fminf(w, 0.f);
        pu += fmaxf(u, 0.f); nu += fminf(u, 0.f);
        pt += fmaxf(t, 0.f); nt += fminf(t, 0.f);
    }
    sc[0 * Bk + b] = pw; sc[1 * Bk + b] = nw;
    sc[2 * Bk + b] = pu; sc[3 * Bk + b] = nu;
    sc[4 * Bk + b] = pt; sc[5 * Bk + b] = nt;
}

// ---------------------------------------------------------------------------
// Kernel 3: rank-6 weight vectors.
//   vecs[0*128+q] = sum_{p: W3[p]>0} W2[q,p]*W3[p]   (A2p)
//   vecs[1*128+q] = sum_{p: W3[p]<=0} W2[q,p]*W3[p]  (A2n)   etc. for W4/W5, W6/W7
// ---------------------------------------------------------------------------
__global__ __launch_bounds__(128) void make_vecs(const float* __restrict__ W2,
                                                 const float* __restrict__ W3,
                                                 const float* __restrict__ W4,
                                                 const float* __restrict__ W5,
                                                 const float* __restrict__ W6,
                                                 const float* __restrict__ W7,
                                                 float* __restrict__ vecs) {
    const int q = threadIdx.x;                           // [0, 128)
    float a2p = 0.f, a2n = 0.f, a4p = 0.f, a4n = 0.f, a6p = 0.f, a6n = 0.f;
    for (int p = 0; p < Pk; ++p) {
        float w3 = W3[p], w2 = W2[q * Pk + p];
        float c2 = w2 * w3;
        if (w3 > 0.f) a2p += c2; else a2n += c2;
        float w5 = W5[p], w4 = W4[q * Pk + p];
        float c4 = w4 * w5;
        if (w5 > 0.f) a4p += c4; else a4n += c4;
        float w7 = W7[p], w6 = W6[q * Pk + p];
        float c6 = w6 * w7;
        if (w7 > 0.f) a6p += c6; else a6n += c6;
    }
    vecs[0 * Pk + q] = a2p; vecs[1 * Pk + q] = a2n;
    vecs[2 * Pk + q] = a4p; vecs[3 * Pk + q] = a4n;
    vecs[4 * Pk + q] = a6p; vecs[5 * Pk + q] = a6n;
}

// ---------------------------------------------------------------------------
// Kernel 4: out = relu( S @ W1^T  +  rank-6 terms  +  xi @ W8^T )
// One block = 256 threads = 8 wave32; block covers 16 rows (b) x 128 cols (q).
// Wave w computes the 16x16 tile at q0 = 16*w via V_WMMA_F32_16X16X4_F32,
// K-loop of 32 steps over p. S tile staged in padded LDS.
//
// Fragment layouts per CDNA5 ISA 7.12.2 (f32, wave32):
//   A 16x4 : lane L<16 -> M=L, v[j]=A[M, j];      lane L>=16 -> M=L-16, v[j]=A[M, 2+j]
//   B 4x16 : lane L<16 -> N=L, v[j]=B[j, N];      lane L>=16 -> N=L-16, v[j]=B[2+j, N]
//   C/D    : lane L<16 -> (M=j,   N=L)   for v[j]; lane L>=16 -> (M=8+j, N=L-16)
// ---------------------------------------------------------------------------
#define LDS_PITCH 132   // 128 + 4 pad -> conflict-free ds_load_b64 fragments

__global__ __launch_bounds__(256) void gemm_wmma(const float* __restrict__ S,
                                                 const float* __restrict__ W1,
                                                 const float* __restrict__ xi,
                                                 const float* __restrict__ W8,
                                                 const float* __restrict__ vecs,
                                                 const float* __restrict__ sc,
                                                 float* __restrict__ out) {
    __shared__ alignas(16) float sS[16 * LDS_PITCH];

    const int tid  = threadIdx.x;
    const int lane = tid & 31;
    const int wave = tid >> 5;
    const int b0   = blockIdx.x * 16;

    // Cooperative staging: 16 rows x 32 float4 = 512 float4, 2 per thread.
#pragma unroll
    for (int f = tid; f < 512; f += 256) {
        const int r  = f >> 5;         // row 0..15
        const int c4 = f & 31;         // float4 col 0..31
        float4 v = *(const float4*)&S[(size_t)(b0 + r) * Pk + c4 * 4];
        float* dst = &sS[r * LDS_PITCH + c4 * 4];
        dst[0] = v.x; dst[1] = v.y; dst[2] = v.z; dst[3] = v.w;
    }
    __syncthreads();

    const int q0   = wave * 16;
    const int half = lane >> 4;        // 0: lanes 0-15, 1: lanes 16-31
    const int l16  = lane & 15;

    v8f acc = {0.f, 0.f, 0.f, 0.f, 0.f, 0.f, 0.f, 0.f};
    const float* w1row = &W1[(size_t)(q0 + l16) * Pk];   // B frag: row q0+l16 of W1
    const float* srow  = &sS[l16 * LDS_PITCH];           // A frag: row b0+l16 of S

#pragma unroll
    for (int ks = 0; ks < 32; ++ks) {
        const int ka = ks * 4 + half * 2;                // K pair owned by this half-wave
        v2f a = *(const v2f*)&srow[ka];                  // ds_load_b64
        v2f b = *(const v2f*)&w1row[ka];                 // global_load_b64 (L2-hot W1)
        acc = __builtin_amdgcn_wmma_f32_16x16x4_f32(
            /*neg_a=*/false, a, /*neg_b=*/false, b,
            /*c_mod=*/(short)0, acc, /*reuse_a=*/false, /*reuse_b=*/false);
    }

    // -------- epilogue: rank-6 terms + xi @ W8^T + relu --------
    const int q = q0 + l16;
    const float a2p = vecs[0 * Pk + q], a2n = vecs[1 * Pk + q];
    const float a4p = vecs[2 * Pk + q], a4n = vecs[3 * Pk + q];
    const float a6p = vecs[4 * Pk + q], a6n = vecs[5 * Pk + q];
    float w8q[7];
#pragma unroll
    for (int j = 0; j < 7; ++j) w8q[j] = W8[q * 7 + j];  // W8 is [P,7] row-major

#pragma unroll
    for (int j = 0; j < 8; ++j) {
        const int b = b0 + j + 8 * half;                 // C/D row for acc[j]
        float e = sc[0 * Bk + b] * a2p + sc[1 * Bk + b] * a2n
                + sc[2 * Bk + b] * a4p + sc[3 * Bk + b] * a4n
                + sc[4 * Bk + b] * a6p + sc[5 * Bk + b] * a6n;
        const float* xr = &xi[b * 7];
#pragma unroll
        for (int t = 0; t < 7; ++t) e += xr[t] * w8q[t];
        const float r = acc[j] + e;
        out[(size_t)b * Pk + q] = fmaxf(r, 0.f);
    }
}

// ---------------------------------------------------------------------------
extern "C" void kernel_launch(void* const* d_in, const int* in_sizes, int n_in,
                              void* d_out, int out_size, void* d_ws, size_t ws_size,
                              hipStream_t stream) {
    const float* xi  = (const float*)d_in[0];
    const float* muN = (const float*)d_in[1];
    const float* wi  = (const float*)d_in[2];
    const float* ui  = (const float*)d_in[3];
    const float* ti  = (const float*)d_in[4];
    const float* W1  = (const float*)d_in[5];
    const float* W2  = (const float*)d_in[6];
    const float* W3  = (const float*)d_in[7];
    const float* W4  = (const float*)d_in[8];
    const float* W5  = (const float*)d_in[9];
    const float* W6  = (const float*)d_in[10];
    const float* W7  = (const float*)d_in[11];
    const float* W8  = (const float*)d_in[12];
    float* out = (float*)d_out;

    char* ws = (char*)d_ws;
    float* S    = (float*)(ws);                                      // B*P floats  (4 MB)
    float* sc   = (float*)(ws + (size_t)Bk * Pk * sizeof(float));    // 6*B floats  (192 KB)
    float* vecs = (float*)(ws + (size_t)Bk * Pk * sizeof(float)
                              + (size_t)6 * Bk * sizeof(float));     // 6*128 floats

    reduce_mu <<<(Bk * Pk / 4) / 256, 256, 0, stream>>>(muN, S);
    reduce_wut<<<Bk / 256,            256, 0, stream>>>(wi, ui, ti, sc);
    make_vecs <<<1,                   128, 0, stream>>>(W2, W3, W4, W5, W6, W7, vecs);
    gemm_wmma <<<Bk / 16,             256, 0, stream>>>(S, W1, xi, W8, vecs, sc, out);
}